// ContinuousEmbedding_25228637896773
// MI455X (gfx1250) — compile-verified
//
#include <hip/hip_runtime.h>

typedef __attribute__((ext_vector_type(2))) float v2f;
typedef __attribute__((ext_vector_type(8))) float v8f;

constexpr int IN   = 256;
constexpr int OUT  = 64;
constexpr int NTOT = IN * OUT;   // 16384
constexpr int BATCH = 4096;

// One wave32 computes a 16(M) x 64(N) tile of out = x @ W  (full GEMM),
// then subtracts the diagonal-block rank-1 term and adds bias:
//   out[m, n] = sum_k x[m,k] W[k,n]  -  x[m, n/64] * W[n/64, n]  +  bias[n]
__global__ __launch_bounds__(256)
void masked_block_gemm_wmma(const float* __restrict__ x,
                            const float* __restrict__ w,
                            const float* __restrict__ bias,
                            float* __restrict__ out)
{
    const int lane = threadIdx.x & 31;
    const int wave = threadIdx.x >> 5;     // 8 waves per block
    const int half = lane >> 4;            // 0: lanes 0-15, 1: lanes 16-31
    const int l16  = lane & 15;

    const int tile = blockIdx.x * 8 + wave;    // 0 .. 65535
    const int n64  = tile & 255;               // N-tile (64 wide) index
    const int mt   = tile >> 8;                // M-tile (16 tall) index

    const int m0 = mt * 16;
    const int n0 = n64 * OUT;

    // A fragment source: row (m0 + l16), K packed 2-per-VGPR:
    //   VGPR0 = K {2*half}, VGPR1 = K {2*half + 1}  -> contiguous b64 load.
    const float* arow = x + (size_t)(m0 + l16) * IN + 2 * half;
    // B fragment source: col (n0 + l16 + 16t), rows k0 + 2*half (+1).
    const float* bcol = w + (size_t)(2 * half) * NTOT + n0 + l16;

    v8f acc0 = {}, acc1 = {}, acc2 = {}, acc3 = {};

#pragma unroll 4
    for (int k0 = 0; k0 < IN; k0 += 4) {
        v2f a = *(const v2f*)(arow + k0);          // global_load_b64
        const float* bp = bcol + (size_t)k0 * NTOT;
        v2f b0, b1, b2, b3;
        b0.x = bp[0];  b0.y = bp[NTOT + 0];
        b1.x = bp[16]; b1.y = bp[NTOT + 16];
        b2.x = bp[32]; b2.y = bp[NTOT + 32];
        b3.x = bp[48]; b3.y = bp[NTOT + 48];
        acc0 = __builtin_amdgcn_wmma_f32_16x16x4_f32(false, a, false, b0, (short)0, acc0, false, false);
        acc1 = __builtin_amdgcn_wmma_f32_16x16x4_f32(false, a, false, b1, (short)0, acc1, false, false);
        acc2 = __builtin_amdgcn_wmma_f32_16x16x4_f32(false, a, false, b2, (short)0, acc2, false, false);
        acc3 = __builtin_amdgcn_wmma_f32_16x16x4_f32(false, a, false, b3, (short)0, acc3, false, false);
    }

    // All 64 columns of this tile belong to diagonal block `n64`.
    const int blk = n64;
    const float* wrow = w + (size_t)blk * NTOT + n0 + l16;
    const float wc0 = wrow[0],  wc1 = wrow[16], wc2 = wrow[32], wc3 = wrow[48];
    const float* brow = bias + n0 + l16;
    const float bs0 = brow[0],  bs1 = brow[16], bs2 = brow[32], bs3 = brow[48];

#pragma unroll
    for (int v = 0; v < 8; ++v) {
        const int m = m0 + v + 8 * half;           // C/D layout: VGPR v, half-wave
        const float xc = x[(size_t)m * IN + blk];  // broadcast load per half-row
        float* orow = out + (size_t)m * NTOT + n0 + l16;
        orow[0]  = acc0[v] - xc * wc0 + bs0;
        orow[16] = acc1[v] - xc * wc1 + bs1;
        orow[32] = acc2[v] - xc * wc2 + bs2;
        orow[48] = acc3[v] - xc * wc3 + bs3;
    }
}

extern "C" void kernel_launch(void* const* d_in, const int* in_sizes, int n_in,
                              void* d_out, int out_size, void* d_ws, size_t ws_size,
                              hipStream_t stream) {
    const float* x    = (const float*)d_in[0];   // [4096, 256]
    const float* w    = (const float*)d_in[1];   // [256, 16384]
    const float* bias = (const float*)d_in[2];   // [16384]
    float* out        = (float*)d_out;           // [4096, 16384] (viewed [4096,256,64])

    const int m_tiles = BATCH / 16;   // 256
    const int n_tiles = NTOT / OUT;   // 256
    const int waves   = m_tiles * n_tiles;       // 65536
    dim3 grid(waves / 8);                        // 8 waves (256 thr) per block
    masked_block_gemm_wmma<<<grid, 256, 0, stream>>>(x, w, bias, out);
}